// SGIntoKGPool_57664230916656
// MI455X (gfx1250) — compile-verified
//
#include <hip/hip_runtime.h>

typedef float v2f __attribute__((ext_vector_type(2)));
typedef float v8f __attribute__((ext_vector_type(8)));
typedef int   v4i __attribute__((ext_vector_type(4)));

// Problem shape (fixed by the reference)
constexpr int BZ = 8;
constexpr int C  = 64;
constexpr int N  = 4096;   // H*W
constexpr int M  = 2048;

// Tiling
constexpr int M_BLK  = 64;          // per workgroup: 4 waves x 16 M-rows
constexpr int K_BLK  = 64;          // n-chunk staged in LDS
constexpr int PITCH  = K_BLK + 4;   // bank stagger (stride 4 banks), rows stay 16B aligned
constexpr int NCHUNK = N / K_BLK;

// ---- CDNA5 async global->LDS probe -------------------------------------
#if defined(__has_builtin)
#if __has_builtin(__builtin_amdgcn_global_load_async_to_lds_b128) && \
    __has_builtin(__builtin_amdgcn_s_wait_asynccnt)
#define USE_ASYNC_LDS 1
#endif
#endif

#define GAS __attribute__((address_space(1)))   // global
#define LAS __attribute__((address_space(3)))   // LDS

__global__ __launch_bounds__(128)
void sg_into_kg_pool(const float* __restrict__ feats,  // (BZ, C, N)
                     const float* __restrict__ adj,    // (BZ, N, M)
                     float* __restrict__ out) {        // (BZ, M, C)
    __shared__ float xs[2][C][PITCH];   // double-buffered X chunk (~34 KB)

    const int tid  = threadIdx.x;
    const int lane = tid & 31;
    const int wave = tid >> 5;
    const int b    = blockIdx.y;
    const int m0   = blockIdx.x * M_BLK + wave * 16;

    const float* __restrict__ xb   = feats + (size_t)b * C * N;
    const float* __restrict__ adjB = adj   + (size_t)b * N * M;

    const int mcol = m0 + (lane & 15);     // this lane's M column (B operand / D column)
    const int koff = (lane >> 4) << 1;     // lanes 0-15: K=0,1 ; lanes 16-31: K=2,3

    v8f acc[4] = {};                       // 16m x 64c in 4 WMMA C/D tiles
    float deg = 0.0f;                      // partial column-sum of adj for this lane's m

    // ---- cooperative X-chunk staging: 64 rows x 64 floats = 1024 float4 ----
    auto stageX = [&](int ci, int buf) {
#ifdef USE_ASYNC_LDS
        // Async DMA global -> LDS (ASYNCcnt): no VGPR staging at all.
#pragma unroll
        for (int p = 0; p < 8; ++p) {
            const int f4 = p * 128 + tid;
            const int c  = f4 >> 4;             // 16 float4 per row
            const int kk = (f4 & 15) << 2;
            GAS v4i* g = (GAS v4i*)(xb + (size_t)c * N + ci * K_BLK + kk);
            LAS v4i* l = (LAS v4i*)&xs[buf][c][kk];
            __builtin_amdgcn_global_load_async_to_lds_b128(g, l, 0, 0);
        }
#else
        // Fallback: register-capped copy (max 4 float4 in flight per batch).
#pragma unroll
        for (int p2 = 0; p2 < 2; ++p2) {
            float4 tmp[4];
#pragma unroll
            for (int q = 0; q < 4; ++q) {
                const int f4 = (p2 * 4 + q) * 128 + tid;
                const int c  = f4 >> 4;
                const int kk = (f4 & 15) << 2;
                tmp[q] = *(const float4*)(xb + (size_t)c * N + ci * K_BLK + kk);
            }
#pragma unroll
            for (int q = 0; q < 4; ++q) {
                const int f4 = (p2 * 4 + q) * 128 + tid;
                const int c  = f4 >> 4;
                const int kk = (f4 & 15) << 2;
                *(float4*)&xs[buf][c][kk] = tmp[q];
            }
            asm volatile("" ::: "memory");      // fence batches: cap staging VGPRs
        }
#endif
    };
    auto waitStage = [&]() {
#ifdef USE_ASYNC_LDS
        __builtin_amdgcn_s_wait_asynccnt(0);
#endif
    };

    stageX(0, 0);
    waitStage();
    __syncthreads();

    int cur = 0;
    for (int ci = 0; ci < NCHUNK; ++ci) {
        if (ci + 1 < NCHUNK) stageX(ci + 1, cur ^ 1);   // overlaps with compute (async path)

        const int n0 = ci * K_BLK;

        // Two half-chunks of 8 K-steps: bf[8] = 16 VGPRs live instead of 32.
#pragma unroll
        for (int h = 0; h < 2; ++h) {
            float2 bf[8];
#pragma unroll
            for (int s = 0; s < 8; ++s) {
                const float* p = adjB + (size_t)(n0 + 32 * h + 4 * s + koff) * M + mcol;
                bf[s].x = p[0];
                bf[s].y = p[M];
            }
#pragma unroll
            for (int s = 0; s < 8; ++s) deg += bf[s].x + bf[s].y;   // free column-degree

            // 4 k-pair groups; each group issues ALL 4 c-tile A-fragment loads
            // (4x ds_load_2addr_b64, clause-able) before its 8 dependent WMMAs,
            // so LDS latency is amortized 4 loads deep instead of 1.
#pragma unroll
            for (int sp = 0; sp < 4; ++sp) {
                const int kb = 32 * h + 8 * sp + koff;
                float2 a[4][2];
#pragma unroll
                for (int t = 0; t < 4; ++t) {
                    const float* r = &xs[cur][t * 16 + (lane & 15)][0];
                    a[t][0] = *(const float2*)(r + kb);
                    a[t][1] = *(const float2*)(r + kb + 4);
                }
#pragma unroll
                for (int j = 0; j < 2; ++j) {
                    const float2 bj = bf[2 * sp + j];
                    const v2f bb = {bj.x, bj.y};
#pragma unroll
                    for (int t = 0; t < 4; ++t) {
                        const v2f av = {a[t][j].x, a[t][j].y};
                        acc[t] = __builtin_amdgcn_wmma_f32_16x16x4_f32(
                            /*neg_a=*/false, av, /*neg_b=*/false, bb,
                            /*c_mod=*/(short)0, acc[t], /*reuse_a=*/false, /*reuse_b=*/false);
                    }
                }
            }
        }

        if (ci + 1 < NCHUNK) {
            waitStage();
            __syncthreads();
            cur ^= 1;
        }
    }

    // Combine the two K-half partial degrees (wave32 shuffle), normalize.
    deg += __shfl_xor(deg, 16);
    const float rnorm = 1.0f / fmaxf(deg, 1.0f);
#pragma unroll
    for (int t = 0; t < 4; ++t)
#pragma unroll
        for (int i = 0; i < 8; ++i) acc[t][i] *= rnorm;

    // Store: lane owns output row m = m0+(lane&15); VGPR v is c-row v (lanes 0-15)
    // or v+8 (lanes 16-31) -> two contiguous float4 stores per c-tile.
    const int    mrow  = m0 + (lane & 15);
    const int    chalf = (lane >> 4) << 3;
    float* __restrict__ ob = out + ((size_t)b * M + mrow) * C + chalf;
#pragma unroll
    for (int t = 0; t < 4; ++t) {
        float4 lo = make_float4(acc[t][0], acc[t][1], acc[t][2], acc[t][3]);
        float4 hi = make_float4(acc[t][4], acc[t][5], acc[t][6], acc[t][7]);
        *(float4*)(ob + t * 16)     = lo;
        *(float4*)(ob + t * 16 + 4) = hi;
    }
}

extern "C" void kernel_launch(void* const* d_in, const int* in_sizes, int n_in,
                              void* d_out, int out_size, void* d_ws, size_t ws_size,
                              hipStream_t stream) {
    const float* feats = (const float*)d_in[0];   // (8, 64, 64, 64) fp32
    const float* adj   = (const float*)d_in[1];   // (8, 4096, 2048) fp32
    float*       out   = (float*)d_out;           // (8, 2048, 64) fp32

    dim3 grid(M / M_BLK, BZ);   // 32 x 8 = 256 workgroups
    dim3 block(128);            // 4 waves (wave32)
    sg_into_kg_pool<<<grid, block, 0, stream>>>(feats, adj, out);
}